// MainNet_51273319579997
// MI455X (gfx1250) — compile-verified
//
#include <hip/hip_runtime.h>
#include <math.h>

// ---------------- constants for the APPM window generator ----------------
#define NW 917
__device__ __constant__ int c_off[14] = {0,121,241,361,442,522,602,651,696,741,789,837,877,917};
__device__ __constant__ int c_rh[13]  = {4,3,5,6,5,7,8,6,10,7,9,7,10};
__device__ __constant__ int c_rw[13]  = {4,5,3,6,7,5,8,10,6,9,7,10,7};
__device__ __constant__ int c_gs[3] = {0, 361, 602};
__device__ __constant__ int c_ge[3] = {361, 602, 917};
__device__ __constant__ int c_np[3] = {3, 2, 1};

typedef __attribute__((ext_vector_type(2))) float v2f;
typedef __attribute__((ext_vector_type(8))) float v8f;

// ---------------- fused APPM: channel-reduce + SAT + scores + NMS --------
// one block per batch element, 256 threads (8 waves)
__global__ void __launch_bounds__(256) appm_kernel(
    const float* __restrict__ fm,     // (256,512,14,14)
    float* __restrict__ out_idx,      // (256,6)  indices as float
    float* __restrict__ out_ps,       // (256,6)  picked scores
    float* __restrict__ out_ws)       // (256,917) all window scores
{
  const int b = blockIdx.x;
  const int t = threadIdx.x;

  __shared__ float s_part[5 * 196];       // 5 channel-group partials
  __shared__ float s_fm[196];
  __shared__ float s_sat[225];            // 15x15 integral image (row0/col0 = 0)
  __shared__ float s_scores[NW];
  __shared__ float s_x0[NW], s_y0[NW], s_x1[NW], s_y1[NW], s_area[NW];
  __shared__ unsigned char s_valid[NW];
  __shared__ float s_rv[256];
  __shared__ int   s_ri[256];
  __shared__ int   s_last;
  __shared__ float s_px0, s_py0, s_px1, s_py1, s_parea;

  // 1) sum over 512 channels -> (14,14); float4 (b128) loads, 16B aligned
  //    since plane stride 196 floats = 49 float4. 5 groups x 49 lanes.
  if (t < 245) {
    const int tg = t / 49;                 // channel group 0..4
    const int s4 = t % 49;                 // float4 spatial index
    float4 acc = make_float4(0.f, 0.f, 0.f, 0.f);
    const float4* p = (const float4*)(fm + (size_t)b * 512 * 196) + s4;
    #pragma unroll 4
    for (int c = tg; c < 512; c += 5) {
      float4 v = p[(size_t)c * 49];
      acc.x += v.x; acc.y += v.y; acc.z += v.z; acc.w += v.w;
    }
    s_part[tg * 196 + 4 * s4 + 0] = acc.x;
    s_part[tg * 196 + 4 * s4 + 1] = acc.y;
    s_part[tg * 196 + 4 * s4 + 2] = acc.z;
    s_part[tg * 196 + 4 * s4 + 3] = acc.w;
  }
  __syncthreads();
  if (t < 196) {
    s_fm[t] = ((s_part[t] + s_part[196 + t]) + (s_part[392 + t] + s_part[588 + t]))
              + s_part[784 + t];
  }
  __syncthreads();

  // 2) 15x15 integral image (padded with a zero row/col)
  if (t < 225) {
    int i = t / 15, j = t % 15;
    s_sat[t] = (i == 0 || j == 0) ? 0.f : s_fm[(i - 1) * 14 + (j - 1)];
  }
  __syncthreads();
  if (t < 15) { float r = 0.f; for (int j = 0; j < 15; ++j) { r += s_sat[t * 15 + j]; s_sat[t * 15 + j] = r; } }
  __syncthreads();
  if (t < 15) { float r = 0.f; for (int i = 0; i < 15; ++i) { r += s_sat[i * 15 + t]; s_sat[i * 15 + t] = r; } }
  __syncthreads();

  // 3) all 917 window scores + box coords
  for (int w = t; w < NW; w += 256) {
    int r = 0;
    while (w >= c_off[r + 1]) ++r;
    const int rh = c_rh[r], rw = c_rw[r];
    const int local = w - c_off[r];
    const int cols = 15 - rw;              // FM - rw + 1
    const int i = local / cols, j = local % cols;
    const float sum = s_sat[(i + rh) * 15 + (j + rw)] - s_sat[i * 15 + (j + rw)]
                    - s_sat[(i + rh) * 15 + j]        + s_sat[i * 15 + j];
    const float score = sum / (float)(rh * rw);
    s_scores[w] = score;
    out_ws[(size_t)b * NW + w] = score;
    const float x0 = (float)(i * 32), y0 = (float)(j * 32);
    const float x1 = x0 + (float)(rh * 32) - 1.f, y1 = y0 + (float)(rw * 32) - 1.f;
    s_x0[w] = x0; s_y0[w] = y0; s_x1[w] = x1; s_y1[w] = y1;
    s_area[w] = (float)(rh * 32) * (float)(rw * 32);
  }
  __syncthreads();

  // 4) per-group NMS (matches jnp: first-index argmax ties, last-idx fallback)
  int slot = 0;
  for (int g = 0; g < 3; ++g) {
    const int gs = c_gs[g], ge = c_ge[g], np = c_np[g];
    for (int w = gs + t; w < ge; w += 256) s_valid[w] = 1;
    if (t == 0) s_last = gs;
    __syncthreads();
    for (int p = 0; p < np; ++p) {
      float bv = -INFINITY; int bi = 0x7fffffff;
      for (int w = gs + t; w < ge; w += 256)
        if (s_valid[w]) { float v = s_scores[w]; if (v > bv) { bv = v; bi = w; } }
      s_rv[t] = bv; s_ri[t] = bi;
      __syncthreads();
      for (int off = 128; off > 0; off >>= 1) {
        if (t < off) {
          float v2 = s_rv[t + off]; int i2 = s_ri[t + off];
          if (v2 > s_rv[t] || (v2 == s_rv[t] && i2 < s_ri[t])) { s_rv[t] = v2; s_ri[t] = i2; }
        }
        __syncthreads();
      }
      if (t == 0) {
        int idx = (s_rv[0] == -INFINITY) ? s_last : s_ri[0];
        s_last = idx;
        out_idx[b * 6 + slot + p] = (float)idx;
        out_ps [b * 6 + slot + p] = s_scores[idx];
        s_px0 = s_x0[idx]; s_py0 = s_y0[idx]; s_px1 = s_x1[idx]; s_py1 = s_y1[idx];
        s_parea = s_area[idx];
      }
      __syncthreads();
      for (int w = gs + t; w < ge; w += 256) {
        float lx = fminf(s_x1[w], s_px1) - fmaxf(s_x0[w], s_px0) + 1.f;
        float ly = fminf(s_y1[w], s_py1) - fmaxf(s_y0[w], s_py0) + 1.f;
        float inter = (lx < 0.f || ly < 0.f) ? 0.f : lx * ly;
        float iou = inter / (s_area[w] + s_parea - inter);
        if (iou > 0.25f) s_valid[w] = 0;
      }
      __syncthreads();
    }
    slot += np;
  }
}

// ---------------- fp32 WMMA GEMM: logits = E @ W_clsT + b ---------------
// one wave per 16x16 tile; K=512 in steps of 4 via v_wmma_f32_16x16x4_f32
__global__ void __launch_bounds__(32) logits_wmma(
    const float* __restrict__ E,   // (256,512)
    const float* __restrict__ Wc,  // (200,512)
    const float* __restrict__ bc,  // (200)
    float* __restrict__ out)       // (256,200)
{
  const int m0 = blockIdx.x * 16;
  const int n0 = blockIdx.y * 16;
  const int l = threadIdx.x;
  const int half = l >> 4, lm = l & 15;
  const int n = n0 + lm;
  const int nc = (n < 200) ? n : 199;          // clamp: D cols are independent
  const float* arow = E  + (size_t)(m0 + lm) * 512 + 2 * half;
  const float* brow = Wc + (size_t)nc * 512 + 2 * half;
  v8f acc = {};
  for (int k0 = 0; k0 < 512; k0 += 4) {
    v2f a = *(const v2f*)(arow + k0);
    v2f bb = *(const v2f*)(brow + k0);
    acc = __builtin_amdgcn_wmma_f32_16x16x4_f32(false, a, false, bb, (short)0, acc, false, false);
  }
  if (n < 200) {
    const int mrow = m0 + half * 8;
    const float bias = bc[n];
    #pragma unroll
    for (int v = 0; v < 8; ++v) out[(size_t)(mrow + v) * 200 + n] = acc[v] + bias;
  }
}

// ---------------- per-row argmax mask: masked = logits * (0.7 + 0.3*oh) --
__global__ void __launch_bounds__(256) mask_kernel(
    const float* __restrict__ logits, float* __restrict__ masked)
{
  const int b = blockIdx.x, t = threadIdx.x;
  __shared__ float rv[256];
  __shared__ int   ri[256];
  float v = (t < 200) ? logits[(size_t)b * 200 + t] : -INFINITY;
  rv[t] = v; ri[t] = (t < 200) ? t : 0x7fffffff;
  __syncthreads();
  for (int off = 128; off > 0; off >>= 1) {
    if (t < off) {
      float v2 = rv[t + off]; int i2 = ri[t + off];
      if (v2 > rv[t] || (v2 == rv[t] && i2 < ri[t])) { rv[t] = v2; ri[t] = i2; }
    }
    __syncthreads();
  }
  const int amax = ri[0];
  if (t < 200) masked[(size_t)b * 200 + t] = v * ((t == amax) ? 1.0f : 0.7f);
}

// ---------------- fp32 WMMA GEMM: hash = tanh(masked @ W_hashT + b) ------
__global__ void __launch_bounds__(32) hash_wmma(
    const float* __restrict__ Mx,  // (256,200)
    const float* __restrict__ Wh,  // (48,200)
    const float* __restrict__ bh,  // (48)
    float* __restrict__ out)       // (256,48)
{
  const int m0 = blockIdx.x * 16;
  const int n0 = blockIdx.y * 16;      // 3 tiles -> N=48 exactly
  const int l = threadIdx.x;
  const int half = l >> 4, lm = l & 15;
  const int j = n0 + lm;
  const float* arow = Mx + (size_t)(m0 + lm) * 200 + 2 * half;
  const float* brow = Wh + (size_t)j * 200 + 2 * half;
  v8f acc = {};
  for (int k0 = 0; k0 < 200; k0 += 4) {
    v2f a = *(const v2f*)(arow + k0);
    v2f bb = *(const v2f*)(brow + k0);
    acc = __builtin_amdgcn_wmma_f32_16x16x4_f32(false, a, false, bb, (short)0, acc, false, false);
  }
  const int mrow = m0 + half * 8;
  const float bias = bh[j];
  #pragma unroll
  for (int v = 0; v < 8; ++v)
    out[(size_t)(mrow + v) * 48 + j] = tanhf(acc[v] + bias);
}

// -------------------------------------------------------------------------
extern "C" void kernel_launch(void* const* d_in, const int* in_sizes, int n_in,
                              void* d_out, int out_size, void* d_ws, size_t ws_size,
                              hipStream_t stream) {
  (void)in_sizes; (void)n_in; (void)out_size; (void)ws_size;
  const float* fm = (const float*)d_in[0];   // (256,512,14,14)
  const float* E  = (const float*)d_in[1];   // (256,512)
  const float* Wc = (const float*)d_in[2];   // (200,512)
  const float* bc = (const float*)d_in[3];   // (200)
  const float* Wh = (const float*)d_in[4];   // (48,200)
  const float* bh = (const float*)d_in[5];   // (48)

  float* out        = (float*)d_out;
  float* out_hash   = out;                       // 256*48
  float* out_logits = out_hash   + 256 * 48;     // 256*200
  float* out_idx    = out_logits + 256 * 200;    // 256*6
  float* out_ps     = out_idx    + 256 * 6;      // 256*6
  float* out_ws     = out_ps     + 256 * 6;      // 256*917

  float* ws_masked = (float*)d_ws;               // 256*200 floats

  // APPM path (independent of GEMM path)
  appm_kernel<<<256, 256, 0, stream>>>(fm, out_idx, out_ps, out_ws);
  // logits -> mask -> hash (stream order enforces dependencies)
  logits_wmma<<<dim3(16, 13), 32, 0, stream>>>(E, Wc, bc, out_logits);
  mask_kernel<<<256, 256, 0, stream>>>(out_logits, ws_masked);
  hash_wmma<<<dim3(16, 3), 32, 0, stream>>>(ws_masked, Wh, bh, out_hash);
}